// AA_GAT_75960791598025
// MI455X (gfx1250) — compile-verified
//
#include <hip/hip_runtime.h>
#include <math.h>

typedef __attribute__((ext_vector_type(2))) float v2f;
typedef __attribute__((ext_vector_type(8))) float v8f;

#define LN_EPS   1e-5f
#define SLOPE    0.01f

// ---------- helpers ----------

__device__ __forceinline__ float eluf(float x) {
    return x > 0.f ? x : (expf(x) - 1.f);
}

// order-preserving float<->uint mapping for atomicMax on floats
__device__ __forceinline__ unsigned f2key(float f) {
    unsigned u = __float_as_uint(f);
    return (u & 0x80000000u) ? ~u : (u | 0x80000000u);
}
__device__ __forceinline__ float key2f(unsigned k) {
    unsigned u = (k & 0x80000000u) ? (k ^ 0x80000000u) : ~k;
    return __uint_as_float(u);
}

// One 16x16 f32 WMMA output tile, accumulating over K (multiple of 4).
// Arow: per-lane pointer to A row (row = lane&15, clamped by caller), stride 1 in K.
// B: column-major-ish row-major [K x ldb], this wave covers cols [0,16) of B.
// V_WMMA_F32_16X16X4_F32: A 16x4, lanes0-15 K={0,1}, lanes16-31 K={2,3};
// B 4x16 mirrored; C/D 8 VGPRs.
__device__ __forceinline__ v8f wmma_accum_f32(const float* __restrict__ Arow,
                                              const float* __restrict__ B,
                                              int ldb, int K, v8f c) {
    const int lane = threadIdx.x & 31;
    const int r    = lane & 15;          // A row / B col handled by this lane
    const int kh   = (lane >> 4) << 1;   // 0 for lanes0-15, 2 for lanes16-31
    for (int k = 0; k < K; k += 4) {
        v2f a, b;
        a.x = Arow[k + kh];
        a.y = Arow[k + kh + 1];
        b.x = B[(size_t)(k + kh) * ldb + r];
        b.y = B[(size_t)(k + kh + 1) * ldb + r];
        c = __builtin_amdgcn_wmma_f32_16x16x4_f32(false, a, false, b,
                                                  (short)0, c, false, false);
    }
    return c;
}

// ---------- zero / init ----------

__global__ void k_zero_f(float* p, size_t n) {
    size_t i = (size_t)blockIdx.x * blockDim.x + threadIdx.x;
    if (i < n) p[i] = 0.f;
}
__global__ void k_zero_u(unsigned* p, int n) {
    int i = blockIdx.x * blockDim.x + threadIdx.x;
    if (i < n) p[i] = 0u;
}

// ---------- node MLP: x = relu(LN(X @ w_node + b)) ----------
// one wave per 16-row tile; WMMA 16x16x4 over K=16, 4 N-tiles of 16 cols
__global__ void k_node_mlp(const float* __restrict__ X, const float* __restrict__ W,
                           const float* __restrict__ b, const float* __restrict__ g,
                           const float* __restrict__ beta,
                           float* __restrict__ xout, int N) {
    __shared__ float sh[16 * 64];
    const int tile = blockIdx.x;
    const int lane = threadIdx.x;
    const int r    = lane & 15;
    int rowg = tile * 16 + r; if (rowg > N - 1) rowg = N - 1;
    const float* Arow = X + (size_t)rowg * 16;

    for (int nt = 0; nt < 4; ++nt) {
        v8f c = {};
        c = wmma_accum_f32(Arow, W + nt * 16, 64, 16, c);
        const int col   = (lane & 15) + nt * 16;
        const int mbase = (lane >> 4) * 8;
        #pragma unroll
        for (int i = 0; i < 8; ++i)
            sh[(mbase + i) * 64 + col] = c[i] + b[col];
    }
    __syncthreads();

    if (lane < 16) {
        int n = tile * 16 + lane;
        if (n < N) {
            float mu = 0.f;
            #pragma unroll
            for (int j = 0; j < 64; ++j) mu += sh[lane * 64 + j];
            mu *= (1.f / 64.f);
            float var = 0.f;
            #pragma unroll
            for (int j = 0; j < 64; ++j) { float d = sh[lane * 64 + j] - mu; var += d * d; }
            var *= (1.f / 64.f);
            float inv = rsqrtf(var + LN_EPS);
            #pragma unroll
            for (int j = 0; j < 64; ++j) {
                float y = (sh[lane * 64 + j] - mu) * inv * g[j] + beta[j];
                xout[(size_t)n * 64 + j] = fmaxf(y, 0.f);
            }
        }
    }
}

// ---------- fused edge MLP + 9 edge projections ----------
// ea = relu(LN(edge_attr @ w_edge + b)); eproj[e][j] = ea . a_edge_j
// (8 head a_edge vectors + final-layer a_edge).  ea never hits memory.
__global__ void k_edge_mlp(const float* __restrict__ edge_attr,
                           const float* __restrict__ w_edge, const float* __restrict__ b_edge,
                           const float* __restrict__ g_edge, const float* __restrict__ beta_edge,
                           const float* __restrict__ gat_a, const float* __restrict__ out_a,
                           float* __restrict__ eproj, int E) {
    __shared__ float sW[8 * 64];
    __shared__ float sb[64], sg[64], sbe[64];
    __shared__ float sA[9 * 64];
    const int t = threadIdx.x;
    for (int i = t; i < 8 * 64; i += blockDim.x) sW[i] = w_edge[i];
    for (int i = t; i < 64; i += blockDim.x) { sb[i] = b_edge[i]; sg[i] = g_edge[i]; sbe[i] = beta_edge[i]; }
    for (int i = t; i < 9 * 64; i += blockDim.x) {
        int hh = i >> 6, j = i & 63;
        sA[i] = (hh < 8) ? gat_a[hh * 192 + 128 + j] : out_a[128 + j];
    }
    __syncthreads();

    int e = blockIdx.x * blockDim.x + t;
    if (e >= E) return;

    float ain[8];
    #pragma unroll
    for (int k = 0; k < 8; ++k) ain[k] = edge_attr[(size_t)e * 8 + k];

    float y[64];
    float mu = 0.f;
    #pragma unroll
    for (int j = 0; j < 64; ++j) {
        float acc = sb[j];
        #pragma unroll
        for (int k = 0; k < 8; ++k) acc += ain[k] * sW[k * 64 + j];
        y[j] = acc; mu += acc;
    }
    mu *= (1.f / 64.f);
    float var = 0.f;
    #pragma unroll
    for (int j = 0; j < 64; ++j) { float d = y[j] - mu; var += d * d; }
    var *= (1.f / 64.f);
    float inv = rsqrtf(var + LN_EPS);

    float acc9[9] = {0.f,0.f,0.f,0.f,0.f,0.f,0.f,0.f,0.f};
    #pragma unroll
    for (int j = 0; j < 64; ++j) {
        float v = fmaxf((y[j] - mu) * inv * sg[j] + sbe[j], 0.f);
        #pragma unroll
        for (int hh = 0; hh < 9; ++hh) acc9[hh] += v * sA[hh * 64 + j];
    }
    #pragma unroll
    for (int hh = 0; hh < 9; ++hh) eproj[(size_t)e * 9 + hh] = acc9[hh];
}

// ---------- per-head h = x @ gat_W[head]; also hs = h.a_src, ht = h.a_tgt ----------
__global__ void k_h_gemm(const float* __restrict__ x, const float* __restrict__ gatW,
                         const float* __restrict__ gat_a,
                         float* __restrict__ h, float* __restrict__ hs, float* __restrict__ ht,
                         int N) {
    __shared__ float sh[16 * 64];
    const int tile = blockIdx.x, head = blockIdx.y;
    const int lane = threadIdx.x;
    const int r    = lane & 15;
    int rowg = tile * 16 + r; if (rowg > N - 1) rowg = N - 1;
    const float* Arow = x + (size_t)rowg * 64;
    const float* B    = gatW + (size_t)head * 64 * 64;

    for (int nt = 0; nt < 4; ++nt) {
        v8f c = {};
        c = wmma_accum_f32(Arow, B + nt * 16, 64, 64, c);
        const int col   = (lane & 15) + nt * 16;
        const int mbase = (lane >> 4) * 8;
        #pragma unroll
        for (int i = 0; i < 8; ++i) sh[(mbase + i) * 64 + col] = c[i];
    }
    __syncthreads();

    const float* a_src = gat_a + (size_t)head * 192;
    const float* a_tgt = a_src + 64;
    if (lane < 16) {
        int n = tile * 16 + lane;
        if (n < N) {
            float ds = 0.f, dt = 0.f;
            float* hrow = h + ((size_t)n * 8 + head) * 64;
            #pragma unroll
            for (int j = 0; j < 64; ++j) {
                float v = sh[lane * 64 + j];
                hrow[j] = v;
                ds += v * a_src[j];
                dt += v * a_tgt[j];
            }
            hs[(size_t)n * 8 + head] = ds;
            ht[(size_t)n * 8 + head] = dt;
        }
    }
}

// ---------- per-edge scores (8 heads) + global max (ordered-uint atomicMax) ----------
__global__ void k_score8(const float* __restrict__ hs, const float* __restrict__ ht,
                         const float* __restrict__ eproj,
                         const int* __restrict__ src, const int* __restrict__ tgt,
                         float* __restrict__ sbuf, unsigned* __restrict__ smax, int E) {
    __shared__ unsigned red[8];
    const int t = threadIdx.x;
    if (t < 8) red[t] = 0u;
    __syncthreads();

    int e = blockIdx.x * blockDim.x + t;
    unsigned keys[8] = {0u,0u,0u,0u,0u,0u,0u,0u};
    if (e < E) {
        int sN = src[e], tN = tgt[e];
        #pragma unroll
        for (int hh = 0; hh < 8; ++hh) {
            float s = hs[(size_t)sN * 8 + hh] + ht[(size_t)tN * 8 + hh]
                    + eproj[(size_t)e * 9 + hh];
            s = s > 0.f ? s : SLOPE * s;
            sbuf[(size_t)e * 8 + hh] = s;
            keys[hh] = f2key(s);
        }
    }
    #pragma unroll
    for (int hh = 0; hh < 8; ++hh) {
        unsigned k = keys[hh];
        for (int off = 16; off; off >>= 1) {
            unsigned o = (unsigned)__shfl_xor((int)k, off, 32);
            k = k > o ? k : o;
        }
        if ((t & 31) == 0) atomicMax(&red[hh], k);
    }
    __syncthreads();
    if (t < 8) atomicMax(&smax[t], red[t]);
}

// ---------- final-layer score (1 "head") ----------
__global__ void k_score1(const float* __restrict__ hs2, const float* __restrict__ ht2,
                         const float* __restrict__ eproj,
                         const int* __restrict__ src, const int* __restrict__ tgt,
                         float* __restrict__ sbuf2, unsigned* __restrict__ smax8, int E) {
    __shared__ unsigned red;
    if (threadIdx.x == 0) red = 0u;
    __syncthreads();
    int e = blockIdx.x * blockDim.x + threadIdx.x;
    unsigned key = 0u;
    if (e < E) {
        float s = hs2[src[e]] + ht2[tgt[e]] + eproj[(size_t)e * 9 + 8];
        s = s > 0.f ? s : SLOPE * s;
        sbuf2[e] = s;
        key = f2key(s);
    }
    for (int off = 16; off; off >>= 1) {
        unsigned o = (unsigned)__shfl_xor((int)key, off, 32);
        key = key > o ? key : o;
    }
    if ((threadIdx.x & 31) == 0) atomicMax(&red, key);
    __syncthreads();
    if (threadIdx.x == 0) atomicMax(smax8, red);
}

// ---------- exp(s - max) and denom segment-sum; works for H=8 or H=1 ----------
__global__ void k_expdenom(const int* __restrict__ src, float* __restrict__ sbuf,
                           const unsigned* __restrict__ smaxk, float* __restrict__ denom,
                           int E, int H) {
    int e = blockIdx.x * blockDim.x + threadIdx.x;
    if (e >= E) return;
    int sN = src[e];
    for (int hh = 0; hh < H; ++hh) {
        float m = key2f(smaxk[hh]);
        float w = expf(sbuf[(size_t)e * H + hh] - m);
        sbuf[(size_t)e * H + hh] = w;
        atomicAdd(&denom[(size_t)sN * H + hh], w);
    }
}

// ---------- hprime[src] += attn * h[tgt] ; block = 1 edge, 8 warps = 8 heads ----------
__global__ void k_aggregate8(const int* __restrict__ src, const int* __restrict__ tgt,
                             const float* __restrict__ ebuf, const float* __restrict__ denom,
                             const float* __restrict__ h, float* __restrict__ hprime, int E) {
    const int e = blockIdx.x;
    const int wid  = threadIdx.x >> 5;   // head
    const int lane = threadIdx.x & 31;
    const int sN = src[e], tN = tgt[e];
    const float attn = ebuf[(size_t)e * 8 + wid] / (denom[(size_t)sN * 8 + wid] + 1e-16f);
    const float* hr = h + ((size_t)tN * 8 + wid) * 64;
    float* hp = hprime + ((size_t)sN * 8 + wid) * 64;
    atomicAdd(&hp[lane],      hr[lane]      * attn);
    atomicAdd(&hp[lane + 32], hr[lane + 32] * attn);
}

__global__ void k_aggregate1(const int* __restrict__ src, const int* __restrict__ tgt,
                             const float* __restrict__ ebuf, const float* __restrict__ denom,
                             const float* __restrict__ h2, float* __restrict__ hprime2, int E) {
    const int e = blockIdx.x;
    const int t = threadIdx.x; // 64 threads
    const int sN = src[e], tN = tgt[e];
    const float attn = ebuf[e] / (denom[sN] + 1e-16f);
    atomicAdd(&hprime2[(size_t)sN * 64 + t], h2[(size_t)tN * 64 + t] * attn);
}

// ---------- xh = elu(elu(hprime)) in place ----------
__global__ void k_elu2(float* __restrict__ v, size_t n) {
    size_t i = (size_t)blockIdx.x * blockDim.x + threadIdx.x;
    if (i >= n) return;
    v[i] = eluf(eluf(v[i]));
}

// ---------- h2 = xh @ out_W  (K=512); also hs2/ht2 dots ----------
__global__ void k_out_gemm(const float* __restrict__ xh, const float* __restrict__ outW,
                           const float* __restrict__ out_a,
                           float* __restrict__ h2, float* __restrict__ hs2,
                           float* __restrict__ ht2, int N) {
    __shared__ float sh[16 * 64];
    const int tile = blockIdx.x;
    const int lane = threadIdx.x;
    const int r    = lane & 15;
    int rowg = tile * 16 + r; if (rowg > N - 1) rowg = N - 1;
    const float* Arow = xh + (size_t)rowg * 512;

    for (int nt = 0; nt < 4; ++nt) {
        v8f c = {};
        c = wmma_accum_f32(Arow, outW + nt * 16, 64, 512, c);
        const int col   = (lane & 15) + nt * 16;
        const int mbase = (lane >> 4) * 8;
        #pragma unroll
        for (int i = 0; i < 8; ++i) sh[(mbase + i) * 64 + col] = c[i];
    }
    __syncthreads();

    if (lane < 16) {
        int n = tile * 16 + lane;
        if (n < N) {
            float ds = 0.f, dt = 0.f;
            float* hrow = h2 + (size_t)n * 64;
            #pragma unroll
            for (int j = 0; j < 64; ++j) {
                float v = sh[lane * 64 + j];
                hrow[j] = v;
                ds += v * out_a[j];
                dt += v * out_a[64 + j];
            }
            hs2[n] = ds;
            ht2[n] = dt;
        }
    }
}

// ---------- out = log_softmax(elu(hprime2)); one wave per node ----------
__global__ void k_final(const float* __restrict__ hprime2, float* __restrict__ out, int N) {
    const int node = blockIdx.x * 8 + (threadIdx.x >> 5);
    const int lane = threadIdx.x & 31;
    if (node >= N) return;
    const float* rr = hprime2 + (size_t)node * 64;
    float y0 = eluf(rr[lane]), y1 = eluf(rr[lane + 32]);
    float m = fmaxf(y0, y1);
    for (int off = 16; off; off >>= 1) m = fmaxf(m, __shfl_xor(m, off, 32));
    float s = expf(y0 - m) + expf(y1 - m);
    for (int off = 16; off; off >>= 1) s += __shfl_xor(s, off, 32);
    float lse = m + logf(s);
    out[(size_t)node * 64 + lane]      = y0 - lse;
    out[(size_t)node * 64 + lane + 32] = y1 - lse;
}

// ---------- host ----------

extern "C" void kernel_launch(void* const* d_in, const int* in_sizes, int n_in,
                              void* d_out, int out_size, void* d_ws, size_t ws_size,
                              hipStream_t stream) {
    const float* X         = (const float*)d_in[0];
    const float* edge_attr = (const float*)d_in[1];
    const int*   eidx      = (const int*)  d_in[2];
    // d_in[3] matched_car_infra_nodes: unused by the reference
    const float* w_node    = (const float*)d_in[4];
    const float* b_node    = (const float*)d_in[5];
    const float* g_node    = (const float*)d_in[6];
    const float* beta_node = (const float*)d_in[7];
    const float* w_edge    = (const float*)d_in[8];
    const float* b_edge    = (const float*)d_in[9];
    const float* g_edge    = (const float*)d_in[10];
    const float* beta_edge = (const float*)d_in[11];
    const float* gat_W     = (const float*)d_in[12];
    const float* gat_a     = (const float*)d_in[13];
    const float* out_W     = (const float*)d_in[14];
    const float* out_a     = (const float*)d_in[15];

    const int N = in_sizes[0] / 16;
    const int E = in_sizes[1] / 8;
    const int* src = eidx;
    const int* tgt = eidx + E;
    const int tiles = (N + 15) / 16;

    // workspace carve (aliased where lifetimes permit)
    char* base = (char*)d_ws;
    size_t off = 0;
    auto carve = [&](size_t bytes) -> char* {
        char* p = base + off;
        off = (off + bytes + 255) & ~(size_t)255;
        return p;
    };
    float*    x      = (float*)   carve((size_t)N * 64 * 4);       // later: hprime2
    float*    h      = (float*)   carve((size_t)N * 8 * 64 * 4);   // later: h2
    float*    hs     = (float*)   carve((size_t)N * 8 * 4);        // later: hs2
    float*    ht     = (float*)   carve((size_t)N * 8 * 4);        // later: ht2
    float*    eproj  = (float*)   carve((size_t)E * 9 * 4);
    float*    sbuf   = (float*)   carve((size_t)E * 8 * 4);        // later: sbuf2
    float*    denom  = (float*)   carve((size_t)N * 8 * 4);        // later: denom2
    float*    hprime = (float*)   carve((size_t)N * 8 * 64 * 4);   // == xh [N,512]
    unsigned* smax   = (unsigned*)carve(16 * 4);
    // aliases (lifetime-disjoint)
    float* hprime2 = x;
    float* h2      = h;
    float* hs2     = hs;
    float* ht2     = ht;
    float* sbuf2   = sbuf;
    float* denom2  = denom;

    // --- init accumulators (every call: graph-replay safe) ---
    {
        size_t n1 = (size_t)N * 8;
        k_zero_f<<<(unsigned)((n1 + 255) / 256), 256, 0, stream>>>(denom, n1);
        size_t n2 = (size_t)N * 512;
        k_zero_f<<<(unsigned)((n2 + 255) / 256), 256, 0, stream>>>(hprime, n2);
        k_zero_u<<<1, 16, 0, stream>>>(smax, 16);
    }

    // --- stage 1: MLPs ---
    k_node_mlp<<<tiles, 32, 0, stream>>>(X, w_node, b_node, g_node, beta_node, x, N);
    k_edge_mlp<<<(E + 127) / 128, 128, 0, stream>>>(edge_attr, w_edge, b_edge, g_edge,
                                                    beta_edge, gat_a, out_a, eproj, E);

    // --- stage 2: 8-head GAT ---
    k_h_gemm<<<dim3(tiles, 8), 32, 0, stream>>>(x, gat_W, gat_a, h, hs, ht, N);
    k_score8<<<(E + 255) / 256, 256, 0, stream>>>(hs, ht, eproj, src, tgt, sbuf, smax, E);
    k_expdenom<<<(E + 255) / 256, 256, 0, stream>>>(src, sbuf, smax, denom, E, 8);
    k_aggregate8<<<E, 256, 0, stream>>>(src, tgt, sbuf, denom, h, hprime, E);

    // --- stage 3: elu(elu(concat)) + output GAT layer ---
    {
        size_t n = (size_t)N * 512;
        k_elu2<<<(unsigned)((n + 255) / 256), 256, 0, stream>>>(hprime, n);
    }
    k_out_gemm<<<tiles, 32, 0, stream>>>(hprime, out_W, out_a, h2, hs2, ht2, N);
    {
        size_t n1 = (size_t)N * 64;
        k_zero_f<<<(unsigned)((n1 + 255) / 256), 256, 0, stream>>>(hprime2, n1);
        size_t n2 = (size_t)N;
        k_zero_f<<<(unsigned)((n2 + 255) / 256), 256, 0, stream>>>(denom2, n2);
    }
    k_score1<<<(E + 255) / 256, 256, 0, stream>>>(hs2, ht2, eproj, src, tgt, sbuf2, smax + 8, E);
    k_expdenom<<<(E + 255) / 256, 256, 0, stream>>>(src, sbuf2, smax + 8, denom2, E, 1);
    k_aggregate1<<<E, 64, 0, stream>>>(src, tgt, sbuf2, denom2, h2, hprime2, E);

    // --- stage 4: elu + log_softmax ---
    k_final<<<(N + 7) / 8, 256, 0, stream>>>(hprime2, (float*)d_out, N);
}